// SelfAttention_15796889714852
// MI455X (gfx1250) — compile-verified
//
#include <hip/hip_runtime.h>
#include <math.h>

// MI455X / gfx1250, wave32. fp32 WMMA path (V_WMMA_F32_16X16X4_F32):
//   A = 16x4 f32 (2 VGPRs/lane), B = 4x16 f32 (2 VGPRs/lane), C/D = 16x16 f32 (8 VGPRs/lane)
typedef __attribute__((ext_vector_type(2))) float v2f;
typedef __attribute__((ext_vector_type(8))) float v8f;

#define BATCH   2
#define SEQ     2048
#define HID     1024
#define NHEAD   16
#define HDIM    64
#define STRIDE_ 128
#define NEGMASK (-10000.0f)

// ---------------------------------------------------------------------------
// Kernel 1: fused Q/V projection.  One wave computes a 32(row) x 32(col) strip
// of BOTH Q and V: two A row-tiles share every B fragment, so each B load
// feeds 2 WMMAs (1 load : 1 WMMA issue ratio) and weight re-streaming traffic
// from L2 is halved vs an M=16 tiling.  8 f32 WMMA accumulators (64 VGPRs).
// Output written head-major [B, H, S, 64] into workspace.
// ---------------------------------------------------------------------------
__global__ __launch_bounds__(256) void qv_proj(
    const float* __restrict__ hid,
    const float* __restrict__ Wq, const float* __restrict__ bq,
    const float* __restrict__ Wv, const float* __restrict__ bv,
    float* __restrict__ qws, float* __restrict__ vws)
{
  const int lane = threadIdx.x & 31;
  const int wid  = (blockIdx.x * blockDim.x + threadIdx.x) >> 5;
  const int row_tile  = wid >> 5;   // 128 tiles of 32 rows (B*S = 4096)
  const int col_strip = wid & 31;   // 32 strips of 32 cols (HID = 1024)
  const int row0 = row_tile << 5;
  const int n0   = col_strip << 5;
  const int lr = lane & 15;         // M index (A) / N index (B,C,D)
  const int lh = lane >> 4;         // lane-half selector

  v8f accq[4] = {{}, {}, {}, {}};   // [rt*2 + nt]
  v8f accv[4] = {{}, {}, {}, {}};

  const float* arow0 = hid + (size_t)(row0 + lr) * HID;        // A tile 0
  const float* arow1 = hid + (size_t)(row0 + 16 + lr) * HID;   // A tile 1
  const float* wqp   = Wq + n0 + lr;                            // B: col = lane&15
  const float* wvp   = Wv + n0 + lr;

  for (int kk = 0; kk < HID; kk += 4) {
    const int ka = kk + lh * 2;                   // K pair owned by this half
    v2f a0, a1;
    a0.x = arow0[ka];  a0.y = arow0[ka + 1];      // contiguous -> b64 load
    a1.x = arow1[ka];  a1.y = arow1[ka + 1];

    const float* wq0 = wqp + (size_t)ka * HID;
    const float* wv0 = wvp + (size_t)ka * HID;
    v2f bq0, bq1, bv0_, bv1_;
    bq0.x  = wq0[0];   bq0.y  = wq0[HID];         // (K,N) and (K+1,N)
    bq1.x  = wq0[16];  bq1.y  = wq0[HID + 16];
    bv0_.x = wv0[0];   bv0_.y = wv0[HID];
    bv1_.x = wv0[16];  bv1_.y = wv0[HID + 16];

    accq[0] = __builtin_amdgcn_wmma_f32_16x16x4_f32(false, a0, false, bq0,  (short)0, accq[0], false, false);
    accq[1] = __builtin_amdgcn_wmma_f32_16x16x4_f32(false, a0, false, bq1,  (short)0, accq[1], false, false);
    accq[2] = __builtin_amdgcn_wmma_f32_16x16x4_f32(false, a1, false, bq0,  (short)0, accq[2], false, false);
    accq[3] = __builtin_amdgcn_wmma_f32_16x16x4_f32(false, a1, false, bq1,  (short)0, accq[3], false, false);
    accv[0] = __builtin_amdgcn_wmma_f32_16x16x4_f32(false, a0, false, bv0_, (short)0, accv[0], false, false);
    accv[1] = __builtin_amdgcn_wmma_f32_16x16x4_f32(false, a0, false, bv1_, (short)0, accv[1], false, false);
    accv[2] = __builtin_amdgcn_wmma_f32_16x16x4_f32(false, a1, false, bv0_, (short)0, accv[2], false, false);
    accv[3] = __builtin_amdgcn_wmma_f32_16x16x4_f32(false, a1, false, bv1_, (short)0, accv[3], false, false);
  }

  const float biasq[2] = { bq[n0 + lr], bq[n0 + 16 + lr] };
  const float biasv[2] = { bv[n0 + lr], bv[n0 + 16 + lr] };

  #pragma unroll
  for (int rt = 0; rt < 2; ++rt) {
    #pragma unroll
    for (int r = 0; r < 8; ++r) {
      const int g = row0 + rt * 16 + r + lh * 8;  // C/D layout: M = r + 8*half
      const int b = g >> 11, s = g & (SEQ - 1);
      #pragma unroll
      for (int nt = 0; nt < 2; ++nt) {
        const int n = n0 + nt * 16 + lr;
        const int h = n >> 6, dd = n & 63;
        const size_t o = (((size_t)(b * NHEAD + h) * SEQ + s) << 6) + dd;
        qws[o] = accq[rt * 2 + nt][r] + biasq[nt];
        vws[o] = accv[rt * 2 + nt][r] + biasv[nt];
      }
    }
  }
}

// ---------------------------------------------------------------------------
// Kernel 2: strided-sparse attention, one wave per (b, h, 16-row query tile).
// Only visits key tiles the mask can admit: local tiles [qt-8, qt] plus
// strided tiles qt-8m (m>=2).  Online (flash) softmax; P converted from C/D
// layout to A layout via a per-wave LDS slab; P@V accumulated with WMMA.
// Fully-masked tiles that we skip contribute exp(~-1250)=0 in fp32 in the
// reference as well, so results match the dense-masked softmax at fp32
// underflow level.
// ---------------------------------------------------------------------------
__global__ __launch_bounds__(256) void sparse_attn(
    const float* __restrict__ qws, const float* __restrict__ vws,
    const float* __restrict__ amask, float* __restrict__ out)
{
  __shared__ float plds[8][16 * 17];   // per-wave 16x16 P tile, stride 17
  const int lane = threadIdx.x & 31;
  const int w    = threadIdx.x >> 5;
  const int wid  = blockIdx.x * 8 + w;
  const int qt = wid & 127;            // query tile (S/16 = 128)
  const int h  = (wid >> 7) & 15;
  const int b  = wid >> 11;
  const int lr = lane & 15;
  const int lh = lane >> 4;
  const int i0 = qt << 4;

  const float* qbase = qws + (size_t)(b * NHEAD + h) * SEQ * HDIM;
  const float* vbase = vws + (size_t)(b * NHEAD + h) * SEQ * HDIM;
  const float* am    = amask + (size_t)b * SEQ;

  // Preload A fragments for the query rows (K = 64 -> 16 steps of 4).
  v2f afrag[16];
  {
    const float* qrow = qbase + (size_t)(i0 + lr) * HDIM;
    #pragma unroll
    for (int t = 0; t < 16; ++t) {
      const int k = t * 4 + lh * 2;
      afrag[t].x = qrow[k];  afrag[t].y = qrow[k + 1];
    }
  }

  v8f o0 = {}, o1 = {}, o2 = {}, o3 = {};
  float mrow[8], lrow[8];
  #pragma unroll
  for (int r = 0; r < 8; ++r) { mrow[r] = -3.0e38f; lrow[r] = 0.0f; }

  auto process_tile = [&](int jt) {
    const int j0 = jt << 4;
    // ---- scores: S = Q_i @ Q_jT  (16 chained f32 WMMAs) ----
    v8f sc = {};
    const float* krow = qbase + (size_t)(j0 + lr) * HDIM;  // B: N = lane&15
    #pragma unroll
    for (int t = 0; t < 16; ++t) {
      const int k = t * 4 + lh * 2;
      v2f bfr;  bfr.x = krow[k];  bfr.y = krow[k + 1];
      sc = __builtin_amdgcn_wmma_f32_16x16x4_f32(false, afrag[t], false, bfr,
                                                 (short)0, sc, false, false);
    }
    // ---- mask, scale, online softmax ----
    const float amj = am[j0 + lr];
    float scale[8];
    #pragma unroll
    for (int r = 0; r < 8; ++r) {
      const int i = i0 + r + lh * 8;
      const int j = j0 + lr;
      const int diff = i - j;
      const bool allowed =
          (diff >= 0) && ((diff < STRIDE_) || ((diff & (STRIDE_ - 1)) == 0));
      float s = (sc[r] + (allowed ? 0.0f : NEGMASK)) * 0.125f + amj;
      float mx = s;
      mx = fmaxf(mx, __shfl_xor(mx, 1, 16));
      mx = fmaxf(mx, __shfl_xor(mx, 2, 16));
      mx = fmaxf(mx, __shfl_xor(mx, 4, 16));
      mx = fmaxf(mx, __shfl_xor(mx, 8, 16));
      const float mnew  = fmaxf(mrow[r], mx);
      const float rescl = expf(mrow[r] - mnew);
      const float pv    = expf(s - mnew);
      float rs = pv;
      rs += __shfl_xor(rs, 1, 16);
      rs += __shfl_xor(rs, 2, 16);
      rs += __shfl_xor(rs, 4, 16);
      rs += __shfl_xor(rs, 8, 16);
      lrow[r]  = lrow[r] * rescl + rs;
      mrow[r]  = mnew;
      scale[r] = rescl;
      plds[w][(r + lh * 8) * 17 + lr] = pv;   // C-layout -> LDS row-major
    }
    #pragma unroll
    for (int r = 0; r < 8; ++r) {
      o0[r] *= scale[r]; o1[r] *= scale[r];
      o2[r] *= scale[r]; o3[r] *= scale[r];
    }
    // ---- P back from LDS in A layout (same-wave DS ops are in-order) ----
    v2f pfr[4];
    #pragma unroll
    for (int t = 0; t < 4; ++t) {
      const int k = t * 4 + lh * 2;
      pfr[t].x = plds[w][lr * 17 + k];
      pfr[t].y = plds[w][lr * 17 + k + 1];
    }
    // ---- O += P @ V  (K = 16 -> 4 steps, 4 d-tiles) ----
    #pragma unroll
    for (int t = 0; t < 4; ++t) {
      const int k = t * 4 + lh * 2;
      const float* vp = vbase + (size_t)(j0 + k) * HDIM + lr;
      v2f vb0, vb1, vb2, vb3;
      vb0.x = vp[0];   vb0.y = vp[HDIM];
      vb1.x = vp[16];  vb1.y = vp[HDIM + 16];
      vb2.x = vp[32];  vb2.y = vp[HDIM + 32];
      vb3.x = vp[48];  vb3.y = vp[HDIM + 48];
      o0 = __builtin_amdgcn_wmma_f32_16x16x4_f32(false, pfr[t], false, vb0, (short)0, o0, false, false);
      o1 = __builtin_amdgcn_wmma_f32_16x16x4_f32(false, pfr[t], false, vb1, (short)0, o1, false, false);
      o2 = __builtin_amdgcn_wmma_f32_16x16x4_f32(false, pfr[t], false, vb2, (short)0, o2, false, false);
      o3 = __builtin_amdgcn_wmma_f32_16x16x4_f32(false, pfr[t], false, vb3, (short)0, o3, false, false);
    }
  };

  // Strided tiles (diagonal-only mask): jt = qt - 8m, m >= 2.
  for (int jt = qt - 16; jt >= 0; jt -= 8) process_tile(jt);
  // Local causal window tiles: [max(0, qt-8), qt].
  const int local_lo = (qt >= 8) ? (qt - 8) : 0;
  for (int jt = local_lo; jt <= qt; ++jt) process_tile(jt);

  // ---- epilogue: divide by row sums, write [B, S, H*d] ----
  #pragma unroll
  for (int r = 0; r < 8; ++r) {
    const float inv = 1.0f / lrow[r];   // diagonal always allowed -> l > 0
    const int i = i0 + r + lh * 8;
    float* op = out + ((size_t)b * SEQ + i) * HID + h * HDIM + lr;
    op[0]  = o0[r] * inv;
    op[16] = o1[r] * inv;
    op[32] = o2[r] * inv;
    op[48] = o3[r] * inv;
  }
}

// ---------------------------------------------------------------------------
extern "C" void kernel_launch(void* const* d_in, const int* in_sizes, int n_in,
                              void* d_out, int out_size, void* d_ws, size_t ws_size,
                              hipStream_t stream) {
  const float* hid   = (const float*)d_in[0];  // [B, S, HID]
  const float* amask = (const float*)d_in[1];  // [B, 1, 1, S]
  const float* Wq    = (const float*)d_in[2];  // [HID, HID]
  const float* bq    = (const float*)d_in[3];  // [HID]
  const float* Wv    = (const float*)d_in[4];  // [HID, HID]
  const float* bv    = (const float*)d_in[5];  // [HID]

  float* qws = (float*)d_ws;                                   // 16 MB
  float* vws = qws + (size_t)BATCH * NHEAD * SEQ * HDIM;       // 16 MB

  // 4096 waves (32-row x 32-col strips) -> 512 blocks of 8 waves
  qv_proj<<<512, 256, 0, stream>>>(hid, Wq, bq, Wv, bv, qws, vws);
  // 4096 waves -> 512 blocks of 8 waves
  sparse_attn<<<512, 256, 0, stream>>>(qws, vws, amask, (float*)d_out);
}